// PredictionHead3D_22290880266839
// MI455X (gfx1250) — compile-verified
//
#include <hip/hip_runtime.h>
#include <math.h>

typedef __attribute__((ext_vector_type(2))) float v2f;
typedef __attribute__((ext_vector_type(8))) float v8f;

#define HWn      6400
#define Wn       80
#define NB       2
#define SEG_THR  0.7f
#define GAUSS_THR 0.7f
#define EPSv     1e-7f
#define PI_L     3.14f
#define NTILES   400      // 6400 pixels / 16
#define CHUNKS   10       // center-tile chunks per pixel tile
#define TPC      40       // center tiles per chunk (400/10)
#define INF_BITS 0x7F800000u

// ---------------------------------------------------------------------------
// Kernel 1: per-center coefficient rows M[b*HW+c][8] and Emin = +inf init.
// M row = [a, -2ax-2by, ax^2+2bxy+cy^2, c, -2cy-2bx, 2b, 0, 0]
// Inactive center -> [0,0,1e30,0,0,0,0,0]  (E = 1e30 -> exp underflows to 0)
// ---------------------------------------------------------------------------
__global__ void prep_kernel(const float* __restrict__ var,
                            const float* __restrict__ seg,
                            float* __restrict__ M,
                            unsigned* __restrict__ Emin) {
    int i = blockIdx.x * blockDim.x + threadIdx.x;
    if (i >= NB * HWn) return;
    Emin[i] = INF_BITS;

    int b = i / HWn, o = i - b * HWn;
    float x = (float)(o / Wn);
    float y = (float)(o % Wn);
    const float* vb = var + (size_t)b * 3 * HWn;
    float v0 = vb[o], v1 = vb[HWn + o], v2 = vb[2 * HWn + o];
    float sg = seg[i];

    float4 lo, hi;
    if (sg > SEG_THR) {
        float vh = fmaxf(v0, 0.f) + 1.f;
        float vw = fmaxf(v1, 0.f) + 1.f;
        float th = PI_L * (1.f / (1.f + expf(-v2)));
        float s = sinf(th), co = cosf(th);
        float vh2 = vh * vh, vw2 = vw * vw;
        float a  = co * co / (2.f * vh2) + s * s / (2.f * vw2);
        float bc = -2.f * s * co / (4.f * vh2) + 2.f * s * co / (4.f * vw2);
        float c  = s * s / (2.f * vh2) + co * co / (2.f * vw2);
        lo = make_float4(a,
                         -(2.f * a * x + 2.f * bc * y),
                         a * x * x + 2.f * bc * x * y + c * y * y,
                         c);
        hi = make_float4(-(2.f * c * y + 2.f * bc * x),
                         2.f * bc, 0.f, 0.f);
    } else {
        lo = make_float4(0.f, 0.f, 1e30f, 0.f);
        hi = make_float4(0.f, 0.f, 0.f, 0.f);
    }
    float* m = M + (size_t)i * 8;
    *(float4*)(m)     = lo;
    *(float4*)(m + 4) = hi;
}

// ---------------------------------------------------------------------------
// Kernel 2: E-tiles via V_WMMA_F32_16X16X4_F32, min-reduced in registers,
// merged across waves with global_atomic_min_u32 (E >= 0 so uint order works).
// One wave per (batch, pixel-tile, center-chunk); grid sized exactly so
// EXEC is all-1s (WMMA requirement).
// ---------------------------------------------------------------------------
__global__ void __launch_bounds__(256)
wmma_min_kernel(const float* __restrict__ M, unsigned* __restrict__ Emin) {
    int wid  = blockIdx.x * (blockDim.x >> 5) + (threadIdx.x >> 5);
    int lane = threadIdx.x & 31;
    int half = lane >> 4;     // 0: lanes 0-15, 1: lanes 16-31
    int l    = lane & 15;

    int chunk = wid % CHUNKS;
    int rest  = wid / CHUNKS;
    int ot    = rest % NTILES;
    int b     = rest / NTILES;

    // B fragment (4x16, K x N): VGPR0 = K(kb), VGPR1 = K(kb+1); kb = half*2.
    // Features F[k] = [p^2, p, 1, q^2, q, pq, 0, 0] computed on the fly.
    int o = ot * 16 + l;
    float fp = (float)(o / Wn), fq = (float)(o % Wn);
    v2f b0, b1;
    if (half) { b0.x = 1.f;     b0.y = fq * fq; b1.x = 0.f; b1.y = 0.f;     }
    else      { b0.x = fp * fp; b0.y = fp;      b1.x = fq;  b1.y = fp * fq; }

    v8f emin;
    for (int r = 0; r < 8; ++r) emin[r] = 3.0e38f;

    const float* Mb = M + (size_t)b * HWn * 8;
    int kb = half * 2;
    int t0 = chunk * TPC;
    for (int t = t0; t < t0 + TPC; ++t) {
        // A fragment (16x4, M x K): lane l holds row t*16+l; VGPR0=K(kb), VGPR1=K(kb+1)
        const float* Mrow = Mb + (size_t)(t * 16 + l) * 8 + kb;
        v2f a0, a1;
        a0.x = Mrow[0]; a0.y = Mrow[1];   // K = kb, kb+1       (first  K-half)
        a1.x = Mrow[4]; a1.y = Mrow[5];   // K = kb+4, kb+5     (second K-half)

        v8f d = {};
        d = __builtin_amdgcn_wmma_f32_16x16x4_f32(false, a0, false, b0,
                                                  (short)0, d, false, false);
        d = __builtin_amdgcn_wmma_f32_16x16x4_f32(false, a1, false, b1,
                                                  (short)0, d, false, false);
        for (int r = 0; r < 8; ++r) emin[r] = fminf(emin[r], d[r]);
    }

    // D layout: VGPR r -> center m=r (lanes 0-15) / m=8+r (lanes 16-31), n=l.
    // Min over this lane's 8 centers, then both halves atomic-min pixel n.
    float mv = emin[0];
    for (int r = 1; r < 8; ++r) mv = fminf(mv, emin[r]);
    mv = fmaxf(mv, 0.0f);   // E >= 0 exactly; guard rounding for uint ordering
    atomicMin(&Emin[b * HWn + ot * 16 + l], __float_as_uint(mv));
}

// ---------------------------------------------------------------------------
// Kernel 3: out = g >= 0.7 ? g : 0, g = exp(-minE + eps). +inf -> 0.
// ---------------------------------------------------------------------------
__global__ void finalize_kernel(const unsigned* __restrict__ Emin,
                                float* __restrict__ out) {
    int i = blockIdx.x * blockDim.x + threadIdx.x;
    if (i >= NB * HWn) return;
    float E = __uint_as_float(Emin[i]);
    float g = expf(-E + EPSv);
    out[i] = (g >= GAUSS_THR) ? g : 0.f;
}

extern "C" void kernel_launch(void* const* d_in, const int* in_sizes, int n_in,
                              void* d_out, int out_size, void* d_ws, size_t ws_size,
                              hipStream_t stream) {
    const float* var = (const float*)d_in[0];   // [2,3,80,80]
    const float* seg = (const float*)d_in[1];   // [2,1,80,80]
    float*    M    = (float*)d_ws;                                   // 2*6400*8 f32 = 400 KB
    unsigned* Emin = (unsigned*)((char*)d_ws +
                                 (size_t)NB * HWn * 8 * sizeof(float)); // 50 KB
    float* out = (float*)d_out;                 // [2,1,80,80]

    prep_kernel<<<(NB * HWn + 255) / 256, 256, 0, stream>>>(var, seg, M, Emin);

    int totalWaves = NB * NTILES * CHUNKS;      // 8000 waves, 8 per 256-thread block
    wmma_min_kernel<<<totalWaves / 8, 256, 0, stream>>>(M, Emin);

    finalize_kernel<<<(NB * HWn + 255) / 256, 256, 0, stream>>>(Emin, out);
}